// PriorWeightPhaseType_67284957659751
// MI455X (gfx1250) — compile-verified
//
#include <hip/hip_runtime.h>

// CDNA5 / gfx1250: wave32, WMMA 16x16 tiles.
// Pack 2 scalars w0,w1 per wave as a block-diagonal 16x16 matrix
// X = diag(w0*S/2^j, w1*S/2^j); expm via order-8 Taylor (Horner, 7 matmuls,
// fully unrolled) + j squarings; every 16x16x16 product = 4 chained
// V_WMMA_F32_16X16X4_F32.
// - X never touches LDS: its A-fragments are S's A-fragments (registers,
//   loaded once), scaled per pair/step by one per-lane scalar.
// - The Horner "+ I" rides for free through the WMMA C operand: the
//   accumulator is seeded with a per-lane identity fragment.
// - Only the evolving T matrix round-trips a wave-private LDS tile
//   (stride 17, bank-conflict free) for the C/D -> A/B layout conversion.

typedef __attribute__((ext_vector_type(2))) float v2f;
typedef __attribute__((ext_vector_type(8))) float v8f;

#define LDS_FENCE() asm volatile("" ::: "memory")

#define STR 17            // padded row stride (dwords) for 16x16 LDS tiles
#define WAVES_PER_BLOCK 8 // 256 threads / wave32

// D = A @ B + C0 with A-fragments in registers (scaled by scaleA) and B in an
// LDS tile (row stride STR). C0 seeds the accumulator (identity for Horner).
__device__ inline v8f mat16_mul_regA(const v2f af[4], const float* B,
                                     float scaleA, v8f c0, int lane) {
  const int m  = lane & 15;   // B col (N) for this lane
  const int hi = lane >> 4;   // lane half selects K pair
  v8f c = c0;
#pragma unroll
  for (int kb = 0; kb < 16; kb += 4) {
    const int ka = kb + 2 * hi;
    v2f a, b;
    a.x = af[kb >> 2].x * scaleA;
    a.y = af[kb >> 2].y * scaleA;
    // B-matrix 4x16 layout (row striped across lanes): N=lane%16, K=2*hi(+1)
    b.x = B[ka * STR + m];
    b.y = B[(ka + 1) * STR + m];
    c = __builtin_amdgcn_wmma_f32_16x16x4_f32(
        /*neg_a=*/false, a, /*neg_b=*/false, b,
        /*c_mod=*/(short)0, c, /*reuse_a=*/false, /*reuse_b=*/false);
  }
  return c;
}

// D = A @ B, both operands from LDS tiles (used for the squaring phase).
__device__ inline v8f mat16_mul(const float* A, const float* B, int lane) {
  const int m  = lane & 15;
  const int hi = lane >> 4;
  v8f c = {};
#pragma unroll
  for (int kb = 0; kb < 16; kb += 4) {
    const int ka = kb + 2 * hi;
    v2f a, b;
    // A-matrix 16x4 layout: VGPR0 holds K=2*hi, VGPR1 holds K=2*hi+1; M=lane%16
    a.x = A[m * STR + ka];
    a.y = A[m * STR + ka + 1];
    b.x = B[ka * STR + m];
    b.y = B[(ka + 1) * STR + m];
    c = __builtin_amdgcn_wmma_f32_16x16x4_f32(
        false, a, false, b, (short)0, c, false, false);
  }
  return c;
}

// Scatter a C/D fragment back to an LDS tile (C/D layout: VGPR r -> row r+8*hi,
// col = lane%16).
__device__ inline void store_cd(float* Dst, v8f c, int lane) {
  const int n  = lane & 15;
  const int hi = lane >> 4;
#pragma unroll
  for (int r = 0; r < 8; ++r) {
    Dst[(r + 8 * hi) * STR + n] = c[r];
  }
}

__global__ __launch_bounds__(256) void pt_expm_kernel(
    const float* __restrict__ w, const float* __restrict__ S,
    const float* __restrict__ alpha, float* __restrict__ partials, int n,
    int nWaves) {
  __shared__ float ldsT[WAVES_PER_BLOCK][16 * STR];
  __shared__ float sS[64];
  __shared__ float sAlpha[8];
  __shared__ float sExit[8];
  __shared__ float sNorm;

  const int tid  = threadIdx.x;
  const int lane = tid & 31;
  const int wid  = tid >> 5;

  if (tid < 64) sS[tid] = S[tid];
  if (tid >= 64 && tid < 72) sAlpha[tid - 64] = alpha[tid - 64];
  __syncthreads();
  if (tid < 8) {
    float r = 0.f;
#pragma unroll
    for (int j = 0; j < 8; ++j) r += sS[tid * 8 + j];
    sExit[tid] = -r;  // exit-rate vector s = -S.sum(axis=1)
  }
  if (tid == 0) {
    float m = 0.f;
    for (int i = 0; i < 8; ++i) {
      float a = 0.f;
      for (int j = 0; j < 8; ++j) a += fabsf(sS[i * 8 + j]);
      m = fmaxf(m, a);
    }
    sNorm = m;  // ||S||_inf
  }
  __syncthreads();

  const int m  = lane & 15;
  const int hi = lane >> 4;

  // Identity fragment in C/D layout: VGPR r holds row r+8*hi, col lane%16.
  v8f idf;
#pragma unroll
  for (int r = 0; r < 8; ++r) idf[r] = ((r + 8 * hi) == m) ? 1.0f : 0.0f;

  // A-layout fragments of the block-diagonal diag(S, S) pattern, held in
  // registers for the whole kernel. Cross-block entries are structurally zero;
  // the per-pair block scale (a0 vs a1) is one per-lane scalar since the
  // block is chosen by the row m alone.
  v2f sfrag[4];
#pragma unroll
  for (int kb = 0; kb < 16; kb += 4) {
    const int ka = kb + 2 * hi;
    float e0 = 0.f, e1 = 0.f;
    if (m < 8) {
      if (ka < 8)     e0 = sS[m * 8 + ka];
      if (ka + 1 < 8) e1 = sS[m * 8 + ka + 1];
    } else {
      if (ka >= 8)     e0 = sS[(m - 8) * 8 + (ka - 8)];
      if (ka + 1 >= 8) e1 = sS[(m - 8) * 8 + (ka + 1 - 8)];
    }
    sfrag[kb >> 2].x = e0;
    sfrag[kb >> 2].y = e1;
  }

  float* T = ldsT[wid];

  const int waveGlobal = blockIdx.x * WAVES_PER_BLOCK + wid;
  const int nPairs     = (n + 1) >> 1;
  float acc            = 0.f;

  for (int p = waveGlobal; p < nPairs; p += nWaves) {
    const int i1    = 2 * p + 1;
    const bool has1 = (i1 < n);
    const float w0  = w[2 * p];
    const float w1  = has1 ? w[i1] : w0;

    // scaling: bring ||X|| <= 0.5 so order-8 Taylor error ~ 0.5^9/9! ~ 5e-9
    const float theta = sNorm * fmaxf(w0, w1);
    int jscale = 0;
    if (theta > 0.5f) {
      jscale = (int)ceilf(log2f(theta * 2.0f));
      jscale = jscale < 0 ? 0 : (jscale > 30 ? 30 : jscale);
    }
    const float sc = exp2f(-(float)jscale);
    const float a0 = w0 * sc, a1 = w1 * sc;
    // per-lane A-fragment scale: row m < 8 -> block 0 (a0), else block 1 (a1)
    const float asel = (m < 8) ? a0 : a1;

    // Build T = I + X/8 (Taylor seed) directly into the LDS tile, no matmul.
#pragma unroll
    for (int r = 0; r < 8; ++r) {
      const int idx = r * 32 + lane;  // 0..255 logical element
      const int row = idx >> 4, col = idx & 15;
      float v = 0.f;
      if (row < 8 && col < 8) v = a0 * sS[row * 8 + col];
      else if (row >= 8 && col >= 8) v = a1 * sS[(row - 8) * 8 + (col - 8)];
      float t = v * 0.125f;
      if (row == col) t += 1.0f;
      T[row * STR + col] = t;
    }
    LDS_FENCE();

    // Horner: for k = 7..1: T = I + (X/k) @ T  => T = sum_{k=0..8} X^k/k!
    // A-fragments from registers (sfrag * asel/k); "+ I" seeds the WMMA C
    // operand; only B (=T) hits LDS.
#pragma unroll
    for (int k = 7; k >= 1; --k) {
      const float rk = asel * (1.0f / (float)k);
      v8f d = mat16_mul_regA(sfrag, T, rk, idf, lane);
      LDS_FENCE();
      store_cd(T, d, lane);
      LDS_FENCE();
    }
    // Undo scaling: T <- T^(2^jscale)
    for (int q = 0; q < jscale; ++q) {
      v8f d = mat16_mul(T, T, lane);
      LDS_FENCE();
      store_cd(T, d, lane);
      LDS_FENCE();
    }

    // density_b = alpha^T * block_b * s ; 64 terms per block, 2 per lane
    float d0 = 0.f, d1 = 0.f;
#pragma unroll
    for (int t = 0; t < 2; ++t) {
      const int e = t * 32 + lane;  // 0..63
      const int i = e >> 3, j = e & 7;
      const float as = sAlpha[i] * sExit[j];
      d0 += as * T[i * STR + j];
      d1 += as * T[(i + 8) * STR + (j + 8)];
    }
#pragma unroll
    for (int off = 16; off > 0; off >>= 1) {
      d0 += __shfl_down(d0, off, 32);
      d1 += __shfl_down(d1, off, 32);
    }
    if (lane == 0) {
      acc += logf(d0);
      if (has1) acc += logf(d1);
    }
  }

  // One deterministic partial per wave (no float atomics).
  if (lane == 0 && waveGlobal < nWaves) partials[waveGlobal] = acc;
}

__global__ __launch_bounds__(256) void pt_reduce_kernel(
    const float* __restrict__ part, int n, float* __restrict__ out) {
  __shared__ float sm[256];
  float s = 0.f;
  for (int i = threadIdx.x; i < n; i += 256) s += part[i];
  sm[threadIdx.x] = s;
  __syncthreads();
  for (int o = 128; o > 0; o >>= 1) {
    if (threadIdx.x < o) sm[threadIdx.x] += sm[threadIdx.x + o];
    __syncthreads();
  }
  if (threadIdx.x == 0) out[0] = sm[0];
}

extern "C" void kernel_launch(void* const* d_in, const int* in_sizes, int n_in,
                              void* d_out, int out_size, void* d_ws,
                              size_t ws_size, hipStream_t stream) {
  const float* w     = (const float*)d_in[0];
  const float* S     = (const float*)d_in[1];
  const float* alpha = (const float*)d_in[2];
  float* out         = (float*)d_out;
  float* partials    = (float*)d_ws;
  const int n        = in_sizes[0];

  const int TPB    = 256;
  int blocks       = 512;
  // keep partials within workspace
  size_t maxWaves  = ws_size / sizeof(float);
  while (blocks > 1 && (size_t)blocks * WAVES_PER_BLOCK > maxWaves) blocks >>= 1;
  const int nWaves = blocks * WAVES_PER_BLOCK;

  pt_expm_kernel<<<blocks, TPB, 0, stream>>>(w, S, alpha, partials, n, nWaves);
  pt_reduce_kernel<<<1, 256, 0, stream>>>(partials, nWaves, out);
}